// PointPillarScatter_Mix_Full_Memory_54211077210395
// MI455X (gfx1250) — compile-verified
//
#include <hip/hip_runtime.h>
#include <hip/hip_bf16.h>

// ---------------------------------------------------------------------------
// PointPillarScatter_Mix_Full_Memory for MI455X (gfx1250, wave32, WMMA).
// fp32 end-to-end using V_WMMA_F32_16X16X4_F32, software-pipelined loads.
// ---------------------------------------------------------------------------

typedef __attribute__((ext_vector_type(2))) float v2f;
typedef __attribute__((ext_vector_type(8))) float v8f;
typedef __attribute__((ext_vector_type(4))) int   v4i_t;

#define NXQ   432
#define NYQ   496
#define NXY   (432 * 496)        // 214272
#define DQ    64
#define KQ    3
#define MQ    1024
#define NVQ   10000
#define NPQ   16384
#define BEVQ  128
#define SHRINKQ 0.0025f

#define AS1Q __attribute__((address_space(1)))
#define AS3Q __attribute__((address_space(3)))

#if defined(__AMDGCN__) && defined(__has_builtin)
#if __has_builtin(__builtin_amdgcn_global_load_async_to_lds_b128) && \
    __has_builtin(__builtin_amdgcn_s_wait_asynccnt)
#define USE_ASYNC_LDS 1
#endif
#endif

static __device__ __forceinline__ v8f wmma_f32(v2f a, v2f b, v8f c) {
  return __builtin_amdgcn_wmma_f32_16x16x4_f32(false, a, false, b, (short)0, c,
                                               false, false);
}

// ---------------------------------------------------------------------------
// K1: per-pillar top-3 nearest points by coordinate distance (pdist2 branch).
// One thread per pillar; point_coords staged through LDS in 16 KB tiles via
// the CDNA5 async global->LDS data mover when available.
// ---------------------------------------------------------------------------
__global__ __launch_bounds__(256, 1)
void k_coord_topk(const int* __restrict__ coords,
                  const float* __restrict__ pcoord,
                  int* __restrict__ idx_c) {
  __shared__ float4 tile[1024];
  const int t = threadIdx.x;
  const int pillar = blockIdx.x * 256 + t;
  float px = 0.f, py = 0.f;
  if (pillar < NVQ) {
    px = (float)coords[pillar * 4 + 1];
    py = (float)coords[pillar * 4 + 2];
  }
  float d0 = 3e38f, d1 = 3e38f, d2 = 3e38f;
  int i0 = 0, i1 = 0, i2 = 0;
  for (int base = 0; base < NPQ; base += 1024) {
#ifdef USE_ASYNC_LDS
    {
      for (int j = t; j < 1024; j += 256) {
        AS1Q v4i_t* g = (AS1Q v4i_t*)(pcoord + (size_t)(base + j) * 4);
        AS3Q v4i_t* l = (AS3Q v4i_t*)(unsigned int)(size_t)&tile[j];
        __builtin_amdgcn_global_load_async_to_lds_b128(g, l, 0, 0);
      }
      __builtin_amdgcn_s_wait_asynccnt(0);
    }
#else
    for (int j = t; j < 1024; j += 256)
      tile[j] = ((const float4*)pcoord)[base + j];
#endif
    __syncthreads();
    if (pillar < NVQ) {
      for (int j = 0; j < 1024; ++j) {
        const float4 p = tile[j];
        const float dx = px - p.y, dy = py - p.z, dz = p.w;
        const float d = dx * dx + dy * dy + dz * dz;
        const int gi = base + j;
        if (d < d2) {
          if (d < d0)      { d2 = d1; i2 = i1; d1 = d0; i1 = i0; d0 = d; i0 = gi; }
          else if (d < d1) { d2 = d1; i2 = i1; d1 = d;  i1 = gi; }
          else             { d2 = d;  i2 = gi; }
        }
      }
    }
    __syncthreads();
  }
  if (pillar < NVQ) {
    idx_c[pillar * 3 + 0] = i0;
    idx_c[pillar * 3 + 1] = i1;
    idx_c[pillar * 3 + 2] = i2;
  }
}

// ---------------------------------------------------------------------------
// K2: S = point_features @ pillar_featuresT, fused per-pillar-column top-3.
// softmax(axis=0) is monotone per column -> top-3 of raw scores == idx_f.
// 313 blocks x 32 pillar columns (two 16-col groups -> halves L2 A-traffic);
// 4 waves stream all 1024 point row-tiles. A fragments ping-pong
// double-buffered; two accumulator chains interleaved per k-step.
// ---------------------------------------------------------------------------
__global__ __launch_bounds__(128, 1)
void k_feat_topk(const float* __restrict__ pf,   // [NP,64]
                 const float* __restrict__ vf,   // [NV,64]
                 int* __restrict__ idx_f) {
  __shared__ float sv[4][32][6];
  __shared__ int   si[4][32][6];
  const int t = threadIdx.x, lane = t & 31, wave = t >> 5;
  const int n = lane & 15, half = lane >> 4;
  const int colbase = blockIdx.x * 32;
  const int c0 = min(colbase + n, NVQ - 1);        // clamp for last block
  const int c1 = min(colbase + 16 + n, NVQ - 1);

  // B frags: B[K][N] = vf[col N][K], lane holds K = 4*ks + 2*half + v
  v2f BfL[16], BfH[16];
  {
    const float* b0 = vf + c0 * DQ + 2 * half;
    const float* b1 = vf + c1 * DQ + 2 * half;
#pragma unroll
    for (int ks = 0; ks < 16; ++ks) {
      BfL[ks] = *(const v2f*)(b0 + ks * 4);
      BfH[ks] = *(const v2f*)(b1 + ks * 4);
    }
  }

  float tA0 = -3e38f, tA1 = -3e38f, tA2 = -3e38f;
  int   iA0 = 0, iA1 = 0, iA2 = 0;
  float tB0 = -3e38f, tB1 = -3e38f, tB2 = -3e38f;
  int   iB0 = 0, iB1 = 0, iB2 = 0;
  const int tiles = NPQ / 16;            // 1024; 256 tiles per wave (even)

  v2f A0[16], A1[16];
  {
    const float* arow = pf + (wave * 16 + n) * DQ + 2 * half;
#pragma unroll
    for (int ks = 0; ks < 16; ++ks) A0[ks] = *(const v2f*)(arow + ks * 4);
  }

  auto top3A = [&](float s, int gi) {
    if (s > tA2) {
      if (s > tA0)      { tA2 = tA1; iA2 = iA1; tA1 = tA0; iA1 = iA0; tA0 = s; iA0 = gi; }
      else if (s > tA1) { tA2 = tA1; iA2 = iA1; tA1 = s;  iA1 = gi; }
      else              { tA2 = s;  iA2 = gi; }
    }
  };
  auto top3B = [&](float s, int gi) {
    if (s > tB2) {
      if (s > tB0)      { tB2 = tB1; iB2 = iB1; tB1 = tB0; iB1 = iB0; tB0 = s; iB0 = gi; }
      else if (s > tB1) { tB2 = tB1; iB2 = iB1; tB1 = s;  iB1 = gi; }
      else              { tB2 = s;  iB2 = gi; }
    }
  };

  auto tileStep = [&](const v2f (&A)[16], int rowbase) {
    v8f acc0 = {}, acc1 = {};
#pragma unroll
    for (int ks = 0; ks < 16; ++ks) {
      acc0 = wmma_f32(A[ks], BfL[ks], acc0);
      acc1 = wmma_f32(A[ks], BfH[ks], acc1);
    }
#pragma unroll
    for (int v = 0; v < 8; ++v) {
      const int gi = rowbase + v + 8 * half;
      top3A(acc0[v], gi);
      top3B(acc1[v], gi);
    }
  };

  for (int rt = wave; rt < tiles; rt += 8) {
    {
      const float* arow = pf + ((rt + 4) * 16 + n) * DQ + 2 * half;
#pragma unroll
      for (int ks = 0; ks < 16; ++ks) A1[ks] = *(const v2f*)(arow + ks * 4);
    }
    tileStep(A0, rt * 16);
    if (rt + 8 < tiles) {
      const float* arow = pf + ((rt + 8) * 16 + n) * DQ + 2 * half;
#pragma unroll
      for (int ks = 0; ks < 16; ++ks) A0[ks] = *(const v2f*)(arow + ks * 4);
    }
    tileStep(A1, (rt + 4) * 16);
  }

  sv[wave][lane][0] = tA0; si[wave][lane][0] = iA0;
  sv[wave][lane][1] = tA1; si[wave][lane][1] = iA1;
  sv[wave][lane][2] = tA2; si[wave][lane][2] = iA2;
  sv[wave][lane][3] = tB0; si[wave][lane][3] = iB0;
  sv[wave][lane][4] = tB1; si[wave][lane][4] = iB1;
  sv[wave][lane][5] = tB2; si[wave][lane][5] = iB2;
  __syncthreads();
  if (t < 32) {
    const int col = colbase + t;
    if (col < NVQ) {
      const int g = t >> 4;       // column group (0: cols 0..15, 1: 16..31)
      const int ln = t & 15;
      float m0 = -3e38f, m1 = -3e38f, m2 = -3e38f;
      int j0 = 0, j1 = 0, j2 = 0;
      for (int w = 0; w < 4; ++w)
        for (int h = 0; h < 2; ++h)
          for (int k = 0; k < 3; ++k) {
            const float s = sv[w][ln + 16 * h][3 * g + k];
            const int gi = si[w][ln + 16 * h][3 * g + k];
            if (s > m2) {
              if (s > m0)      { m2 = m1; j2 = j1; m1 = m0; j1 = j0; m0 = s; j0 = gi; }
              else if (s > m1) { m2 = m1; j2 = j1; m1 = s;  j1 = gi; }
              else             { m2 = s;  j2 = gi; }
            }
          }
      idx_f[col * 3 + 0] = j0;
      idx_f[col * 3 + 1] = j1;
      idx_f[col * 3 + 2] = j2;
    }
  }
}

// ---------------------------------------------------------------------------
// K3: MemAE memory lookup, fully fused per 16-row tile:
//   att = softmax(X @ mem_wT); hard-shrink-relu; L1 renorm; out = att @ mem_w
// X rows gathered on the fly via idx. att tile (16x1024 fp32 = 64 KB) lives
// in LDS; row reductions via wave32 shfl_xor. Both GEMM phases prefetch
// fragment groups of 16 k-steps so waits amortize over 16 WMMAs.
// grid.y selects branch (0: feature idx_f -> out_f, 1: coord idx_c -> out_c).
// ---------------------------------------------------------------------------
__global__ __launch_bounds__(128, 1)
void k_memlookup(const float* __restrict__ pf,     // [NP,64]
                 const int* __restrict__ idx_f,
                 const int* __restrict__ idx_c,
                 const float* __restrict__ mem_w,  // [M,64]
                 float* __restrict__ out_f,        // [NV*K,64]
                 float* __restrict__ out_c) {
  __shared__ float att[16][MQ];
  const int* idx = blockIdx.y ? idx_c : idx_f;
  float* out = blockIdx.y ? out_c : out_f;
  const int t = threadIdx.x, lane = t & 31, wave = t >> 5;
  const int n = lane & 15, half = lane >> 4;
  const int rowbase = blockIdx.x * 16;

  // Gathered A frags: A[M][K] = pf[idx[rowbase+M]][K]
  const float* arow = pf + (long)idx[rowbase + n] * DQ + 2 * half;
  v2f Af[16];
#pragma unroll
  for (int ks = 0; ks < 16; ++ks) Af[ks] = *(const v2f*)(arow + ks * 4);

  // Phase 1: att[16][1024] = X @ mem_wT   (B[K][N] = mem_w[cbase+N][K])
  for (int ct = wave; ct < MQ / 16; ct += 4) {
    const int cbase = ct * 16;
    const float* brow = mem_w + (cbase + n) * DQ + 2 * half;
    v2f Bfr[16];
#pragma unroll
    for (int ks = 0; ks < 16; ++ks) Bfr[ks] = *(const v2f*)(brow + ks * 4);
    v8f acc = {};
#pragma unroll
    for (int ks = 0; ks < 16; ++ks) acc = wmma_f32(Af[ks], Bfr[ks], acc);
#pragma unroll
    for (int v = 0; v < 8; ++v) att[v + 8 * half][cbase + n] = acc[v];
  }
  __syncthreads();

  // softmax + hard_shrink_relu + L1 renorm; 8 lanes per row (wave32 shfl)
  {
    const int row = t >> 3, sub = t & 7;
    float mx = -3e38f;
    for (int j = sub; j < MQ; j += 8) mx = fmaxf(mx, att[row][j]);
    for (int m = 1; m < 8; m <<= 1) mx = fmaxf(mx, __shfl_xor(mx, m, 32));
    float sum = 0.f;
    for (int j = sub; j < MQ; j += 8) {
      const float e = __expf(att[row][j] - mx);
      att[row][j] = e;
      sum += e;
    }
    for (int m = 1; m < 8; m <<= 1) sum += __shfl_xor(sum, m, 32);
    const float inv = 1.0f / sum;
    float l1 = 0.f;
    for (int j = sub; j < MQ; j += 8) {
      const float a = att[row][j] * inv;
      const float sh = fmaxf(a - SHRINKQ, 0.f) * a / (fabsf(a - SHRINKQ) + 1e-12f);
      att[row][j] = sh;
      l1 += fabsf(sh);
    }
    for (int m = 1; m < 8; m <<= 1) l1 += __shfl_xor(l1, m, 32);
    const float rinv = 1.0f / fmaxf(l1, 1e-12f);
    for (int j = sub; j < MQ; j += 8) att[row][j] *= rinv;
  }
  __syncthreads();

  // Phase 2: out = att @ mem_w; wave w owns output d-tile w (4x16=64 cols).
  // 256 k-steps chunked by 16 with prefetched A (LDS) and B (global) frags.
  {
    const int dbase = wave * 16;
    v8f acc = {};
    for (int kb0 = 0; kb0 < MQ; kb0 += 64) {
      v2f Afr[16], Bfr[16];
#pragma unroll
      for (int ks = 0; ks < 16; ++ks) {
        const int kk = kb0 + ks * 4 + 2 * half;
        Afr[ks] = *(const v2f*)&att[n][kk];
        v2f b;                               // B[K][N] = mem_w[kk+K][dbase+N]
        b[0] = mem_w[(kk + 0) * DQ + dbase + n];
        b[1] = mem_w[(kk + 1) * DQ + dbase + n];
        Bfr[ks] = b;
      }
#pragma unroll
      for (int ks = 0; ks < 16; ++ks) acc = wmma_f32(Afr[ks], Bfr[ks], acc);
    }
#pragma unroll
    for (int v = 0; v < 8; ++v)
      out[(rowbase + v + 8 * half) * DQ + dbase + n] = acc[v];
  }
}

// ---------------------------------------------------------------------------
// K4: f_pre / c_pre = out_x[NV,192] @ adapt_wT[192,64]  (grid.y = branch)
// 48 k-steps chunked by 16 with prefetched fragment arrays.
// ---------------------------------------------------------------------------
__global__ __launch_bounds__(128, 1)
void k_adapt(const float* __restrict__ out_f,
             const float* __restrict__ out_c,
             const float* __restrict__ adapt_w,   // [64,192]
             float* __restrict__ f_pre,
             float* __restrict__ c_pre) {
  const float* X = blockIdx.y ? out_c : out_f;
  float* Y = blockIdx.y ? c_pre : f_pre;
  const int t = threadIdx.x, lane = t & 31, wave = t >> 5;
  const int n = lane & 15, half = lane >> 4;
  const int rowbase = blockIdx.x * 16;
  const int dbase = wave * 16;                       // 4 waves cover 64 cols
  const float* arow = X + (rowbase + n) * (KQ * DQ) + 2 * half;
  const float* brow = adapt_w + (dbase + n) * (KQ * DQ) + 2 * half;
  v8f acc = {};
#pragma unroll
  for (int c = 0; c < 3; ++c) {
    v2f Afr[16], Bfr[16];
#pragma unroll
    for (int ks = 0; ks < 16; ++ks) {
      Afr[ks] = *(const v2f*)(arow + (c * 16 + ks) * 4);
      Bfr[ks] = *(const v2f*)(brow + (c * 16 + ks) * 4);
    }
#pragma unroll
    for (int ks = 0; ks < 16; ++ks) acc = wmma_f32(Afr[ks], Bfr[ks], acc);
  }
#pragma unroll
  for (int v = 0; v < 8; ++v)
    Y[(rowbase + v + 8 * half) * DQ + dbase + n] = acc[v];
}

// ---------------------------------------------------------------------------
// K5: w_pre = pillar_features @ weight_wT  ([NV,2], tiny -> plain VALU)
// ---------------------------------------------------------------------------
__global__ __launch_bounds__(128, 1)
void k_weight(const float* __restrict__ vf,
              const float* __restrict__ weight_w,  // [2,64]
              float* __restrict__ w_pre) {
  const int i = blockIdx.x * 128 + threadIdx.x;
  if (i >= NVQ * 2) return;
  const int p = i >> 1, c = i & 1;
  const float* a = vf + p * DQ;
  const float* b = weight_w + c * DQ;
  float s = 0.f;
#pragma unroll
  for (int k = 0; k < DQ; ++k) s += a[k] * b[k];
  w_pre[p * 2 + c] = s;
}

// ---------------------------------------------------------------------------
// K6: BatchNorm training stats (biased var) per column.
// cols 0..63: f_pre, 64..127: c_pre, 128..129: w_pre.
// ---------------------------------------------------------------------------
__global__ __launch_bounds__(256, 1)
void k_bnstats(const float* __restrict__ f_pre,
               const float* __restrict__ c_pre,
               const float* __restrict__ w_pre,
               float* __restrict__ mu, float* __restrict__ rs) {
  __shared__ float ls[256], ls2[256];
  const int col = blockIdx.x;
  const float* X;
  int stride, cc;
  if (col < 64)       { X = f_pre; stride = DQ; cc = col; }
  else if (col < 128) { X = c_pre; stride = DQ; cc = col - 64; }
  else                { X = w_pre; stride = 2;  cc = col - 128; }
  float s = 0.f, s2 = 0.f;
  for (int r = threadIdx.x; r < NVQ; r += 256) {
    const float v = X[r * stride + cc];
    s += v;
    s2 += v * v;
  }
  ls[threadIdx.x] = s;
  ls2[threadIdx.x] = s2;
  __syncthreads();
  for (int o = 128; o > 0; o >>= 1) {
    if (threadIdx.x < o) {
      ls[threadIdx.x] += ls[threadIdx.x + o];
      ls2[threadIdx.x] += ls2[threadIdx.x + o];
    }
    __syncthreads();
  }
  if (threadIdx.x == 0) {
    const float m = ls[0] / (float)NVQ;
    const float var = ls2[0] / (float)NVQ - m * m;
    mu[col] = m;
    rs[col] = rsqrtf(var + 1e-3f);
  }
}

// ---------------------------------------------------------------------------
// K7: zero canvas (131 x 214272 floats), float4 stores.
// ---------------------------------------------------------------------------
__global__ __launch_bounds__(256, 1)
void k_zero(float4* __restrict__ out, int n4) {
  const int i = blockIdx.x * 256 + threadIdx.x;
  if (i < n4) out[i] = make_float4(0.f, 0.f, 0.f, 0.f);
}

// ---------------------------------------------------------------------------
// K8: fused BN+ReLU (both branches), 2-way softmax gate, scatter to BEV.
// One block per pillar, 128 threads = 128 output channels; t<3 also write pind.
// ---------------------------------------------------------------------------
__global__ __launch_bounds__(128, 1)
void k_scatter(const float* __restrict__ vf,
               const int* __restrict__ coords,
               const float* __restrict__ f_pre,
               const float* __restrict__ c_pre,
               const float* __restrict__ w_pre,
               const float* __restrict__ mu,
               const float* __restrict__ rs,
               const float* __restrict__ g1, const float* __restrict__ b1,
               const float* __restrict__ g2, const float* __restrict__ b2,
               float* __restrict__ out) {
  __shared__ float w01[2];
  const int p = blockIdx.x, t = threadIdx.x;
  const int x = coords[p * 4 + 1];
  const int y = coords[p * 4 + 2];
  const int z = coords[p * 4 + 3];
  const int bev = x + y * NXQ + z;
  if (t == 0) {
    const float t0 = (w_pre[p * 2 + 0] - mu[128]) * rs[128] * g2[0] + b2[0];
    const float t1 = (w_pre[p * 2 + 1] - mu[129]) * rs[129] * g2[1] + b2[1];
    const float m = fmaxf(t0, t1);
    const float e0 = __expf(t0 - m), e1 = __expf(t1 - m);
    const float inv = 1.0f / (e0 + e1);
    w01[0] = e0 * inv;
    w01[1] = e1 * inv;
  }
  __syncthreads();
  float val;
  if (t < 64) {
    val = vf[p * DQ + t];
  } else {
    const int o = t - 64;
    const float f = fmaxf((f_pre[p * DQ + o] - mu[o]) * rs[o] * g1[o] + b1[o], 0.f);
    const float c = fmaxf((c_pre[p * DQ + o] - mu[64 + o]) * rs[64 + o] * g1[o] + b1[o], 0.f);
    val = w01[0] * f + w01[1] * c;
  }
  out[(size_t)t * NXY + bev] = val;
  if (t < 3) {
    const float pv = (t == 0) ? (float)y : (t == 1) ? (float)z : (float)x;
    out[(size_t)(BEVQ + t) * NXY + bev] = pv;
  }
}

// ---------------------------------------------------------------------------
// Host launcher
// ---------------------------------------------------------------------------
extern "C" void kernel_launch(void* const* d_in, const int* in_sizes, int n_in,
                              void* d_out, int out_size, void* d_ws, size_t ws_size,
                              hipStream_t stream) {
  (void)in_sizes; (void)n_in; (void)out_size; (void)ws_size;
  const float* pillar_features = (const float*)d_in[0];   // [NV,64]
  const int*   coords          = (const int*)d_in[1];     // [NV,4]
  const float* point_features  = (const float*)d_in[2];   // [NP,64]
  const float* point_coords    = (const float*)d_in[3];   // [NP,4]
  const float* adapt_w         = (const float*)d_in[4];   // [64,192]
  const float* bn1_g           = (const float*)d_in[5];
  const float* bn1_b           = (const float*)d_in[6];
  const float* weight_w        = (const float*)d_in[7];   // [2,64]
  const float* bn2_g           = (const float*)d_in[8];
  const float* bn2_b           = (const float*)d_in[9];
  const float* mem_w           = (const float*)d_in[10];  // [1024,64]
  float* out = (float*)d_out;

  // workspace carve-up (256 B aligned)
  char* ws = (char*)d_ws;
  auto carve = [&](size_t bytes) {
    char* p = ws;
    ws += (bytes + 255) & ~(size_t)255;
    return p;
  };
  int*   idx_c = (int*)carve(sizeof(int) * NVQ * KQ);
  int*   idx_f = (int*)carve(sizeof(int) * NVQ * KQ);
  float* out_f = (float*)carve(sizeof(float) * NVQ * KQ * DQ);
  float* out_c = (float*)carve(sizeof(float) * NVQ * KQ * DQ);
  float* f_pre = (float*)carve(sizeof(float) * NVQ * DQ);
  float* c_pre = (float*)carve(sizeof(float) * NVQ * DQ);
  float* w_pre = (float*)carve(sizeof(float) * NVQ * 2);
  float* mu    = (float*)carve(sizeof(float) * 130);
  float* rs    = (float*)carve(sizeof(float) * 130);

  // 1) coord top-3 (async LDS staging)
  k_coord_topk<<<dim3((NVQ + 255) / 256), dim3(256), 0, stream>>>(
      coords, point_coords, idx_c);
  // 2) feature GEMM + fused top-3 (WMMA f32, double-buffered A, 32-wide cols)
  k_feat_topk<<<dim3((NVQ + 31) / 32), dim3(128), 0, stream>>>(
      point_features, pillar_features, idx_f);
  // 3) gate logits
  k_weight<<<dim3((NVQ * 2 + 127) / 128), dim3(128), 0, stream>>>(
      pillar_features, weight_w, w_pre);
  // 4) fused memory lookup, both branches (WMMA f32 x2 GEMMs per tile)
  k_memlookup<<<dim3((NVQ * KQ) / 16, 2), dim3(128), 0, stream>>>(
      point_features, idx_f, idx_c, mem_w, out_f, out_c);
  // 5) adapt GEMM, both branches (WMMA f32)
  k_adapt<<<dim3(NVQ / 16, 2), dim3(128), 0, stream>>>(
      out_f, out_c, adapt_w, f_pre, c_pre);
  // 6) BN training stats (130 columns)
  k_bnstats<<<dim3(130), dim3(256), 0, stream>>>(f_pre, c_pre, w_pre, mu, rs);
  // 7) zero canvas, then scatter
  const int n4 = ((BEVQ + 3) * NXY) / 4;
  k_zero<<<dim3((n4 + 255) / 256), dim3(256), 0, stream>>>((float4*)out, n4);
  k_scatter<<<dim3(NVQ), dim3(128), 0, stream>>>(
      pillar_features, coords, f_pre, c_pre, w_pre, mu, rs,
      bn1_g, bn1_b, bn2_g, bn2_b, out);
}